// MambaEncoder_1589137900242
// MI455X (gfx1250) — compile-verified
//
#include <hip/hip_runtime.h>
#include <math.h>

// ---------------- problem constants ----------------
#define BB 2
#define CC 192
#define LL 32          // each spatial axis length
#define DI 384         // EXP*C
#define SS 16
#define KK 4
#define RR 12
#define VOL 32768      // 32*32*32

typedef float v2f __attribute__((ext_vector_type(2)));
typedef float v8f __attribute__((ext_vector_type(8)));

// ---------------------------------------------------------------------------
// Block-level GEMM: D(MxN) = A(MxK, LDS row-major) * W^T  (W is (N,K) row-major
// in global memory).  Uses V_WMMA_F32_16X16X4_F32 tiles, one tile per wave,
// wave-uniform tile loop so EXEC is all-ones at every wmma.
// Epilogue functor is called as epi(m, n, value) for in-range elements.
// ---------------------------------------------------------------------------
template <class Epi>
__device__ __forceinline__ void gemm_xwt(const float* __restrict__ sA, int lda,
                                         const float* __restrict__ gW, int ldw,
                                         int M, int N, int K, Epi epi) {
  const int lane   = threadIdx.x & 31;
  const int wave   = threadIdx.x >> 5;
  const int nwaves = (int)(blockDim.x >> 5);
  const int mt = (M + 15) >> 4;
  const int nt = (N + 15) >> 4;
  const int m_in  = lane & 15;          // A-frag row / B-frag col / D col
  const int khalf = (lane >> 4) * 2;    // K sub-offset per lane half

  for (int tile = wave; tile < mt * nt; tile += nwaves) {
    const int m0 = (tile / nt) * 16;
    const int n0 = (tile % nt) * 16;
    v8f acc = {0.f, 0.f, 0.f, 0.f, 0.f, 0.f, 0.f, 0.f};
    const bool nok = (n0 + m_in) < N;
    const float* __restrict__ arow = sA + (size_t)(m0 + m_in) * lda;
    const float* __restrict__ wrow = gW + (size_t)(n0 + m_in) * ldw;
    for (int k0 = 0; k0 < K; k0 += 4) {
      v2f a, b;
      a.x = arow[k0 + khalf];
      a.y = arow[k0 + khalf + 1];
      b.x = nok ? wrow[k0 + khalf]     : 0.f;
      b.y = nok ? wrow[k0 + khalf + 1] : 0.f;
      acc = __builtin_amdgcn_wmma_f32_16x16x4_f32(false, a, false, b,
                                                  (short)0, acc, false, false);
    }
    const int mbase = m0 + (lane >> 4) * 8;
    const int ncol  = n0 + m_in;
    if (ncol < N) {
#pragma unroll
      for (int v = 0; v < 8; ++v) {
        const int m = mbase + v;
        if (m < M) epi(m, ncol, acc[v]);
      }
    }
  }
}

__device__ __forceinline__ float warp_sum(float v) {
#pragma unroll
  for (int off = 16; off > 0; off >>= 1) v += __shfl_xor(v, off, 32);
  return v;
}

// ---------------------------------------------------------------------------
// Kernel 1: one pass over x computing all three axis means.
// grid = B*C blocks of 256 threads; block handles one (b,c) 32^3 slice.
// ---------------------------------------------------------------------------
__global__ void k_axis_means(const float* __restrict__ x, float* __restrict__ seqs) {
  __shared__ float s_d[32], s_h[32], s_w[32];
  const int bc = blockIdx.x;
  const int t = threadIdx.x;
  if (t < 96) {
    if (t < 32) s_d[t] = 0.f; else if (t < 64) s_h[t - 32] = 0.f; else s_w[t - 64] = 0.f;
  }
  __syncthreads();
  const int lane = t & 31, grp = t >> 5;
  const float* __restrict__ px = x + (size_t)bc * VOL;
  float accw = 0.f;
  for (int q = 0; q < 128; ++q) {
    const int p = grp * 128 + q;       // p = d*32 + h
    const int d = p >> 5, h = p & 31;
    const float v = px[(size_t)p * 32 + lane];
    accw += v;
    const float rs = warp_sum(v);
    if (lane == 0) { atomicAdd(&s_h[h], rs); atomicAdd(&s_d[d], rs); }
  }
  atomicAdd(&s_w[lane], accw);
  __syncthreads();
  if (t < 96) {
    const int i = t >> 5, l = t & 31;
    const float v = (i == 0 ? s_d[l] : (i == 1 ? s_h[l] : s_w[l])) * (1.0f / 1024.0f);
    const int b = bc / CC, c = bc % CC;
    seqs[(((size_t)i * BB + b) * CC + c) * LL + l] = v;
  }
}

// ---------------------------------------------------------------------------
// Kernel 2: full Mamba branch per (i, b).  grid = 6 blocks of 256 threads.
// All tensors live in LDS; sequential scan keeps state in VGPRs.
// ---------------------------------------------------------------------------
__global__ void k_branch(const float* __restrict__ seqs,
                         const float* __restrict__ ln_w, const float* __restrict__ ln_b,
                         const float* __restrict__ in_w, const float* __restrict__ conv_w,
                         const float* __restrict__ conv_b, const float* __restrict__ xp_w,
                         const float* __restrict__ dt_w, const float* __restrict__ dt_b,
                         const float* __restrict__ A_log, const float* __restrict__ Dskip,
                         const float* __restrict__ out_w, const float* __restrict__ up_w,
                         const float* __restrict__ up_b,
                         float* __restrict__ Zg, float* __restrict__ Zmean) {
  extern __shared__ float sm[];
  float* s_seq = sm;            // 6144  : seq (32x192), later tok
  float* s_xm  = sm + 6144;     // 12288 : xm (32x384), later dt, later y
  float* s_z   = sm + 18432;    // 12288 : z  (32x384)
  float* s_xc  = sm + 30720;    // 12288 : xc (32x384), later Z (32x192)
  float* s_dbc = sm + 43008;    // 1536  : dbc (32x48, 44 valid)
  float* s_An  = sm + 44544;    // 6144  : -exp(A_log)  [ch*16+s]

  const int i = blockIdx.x >> 1;
  const int b = blockIdx.x & 1;
  const int t = threadIdx.x;
  const int lane = t & 31, wave = t >> 5;

  // load seq (transposed to rows=l, cols=c) and A = -exp(A_log)
  for (int idx = t; idx < 6144; idx += 256) {
    const int l = idx / CC, c = idx % CC;
    s_seq[idx] = seqs[(((size_t)i * BB + b) * CC + c) * LL + l];
    s_An[idx]  = -expf(A_log[(size_t)i * (DI * SS) + idx]);
  }
  __syncthreads();

  // LayerNorm over C per row
  for (int r = wave; r < LL; r += 8) {
    float s = 0.f, ss = 0.f;
    for (int c = lane; c < CC; c += 32) { const float v = s_seq[r * CC + c]; s += v; ss += v * v; }
    s = warp_sum(s); ss = warp_sum(ss);
    const float mu = s * (1.f / CC);
    const float var = ss * (1.f / CC) - mu * mu;
    const float rstd = rsqrtf(var + 1e-5f);
    for (int c = lane; c < CC; c += 32) {
      const float v = s_seq[r * CC + c];
      s_seq[r * CC + c] = (v - mu) * rstd * ln_w[i * CC + c] + ln_b[i * CC + c];
    }
  }
  __syncthreads();

  // GEMM1: (32x192) @ in_w^T (192x768) -> xm | z
  gemm_xwt(s_seq, CC, in_w + (size_t)i * (2 * DI) * CC, CC, LL, 2 * DI, CC,
           [&](int m, int n, float v) {
             if (n < DI) s_xm[m * DI + n] = v; else s_z[m * DI + (n - DI)] = v;
           });
  __syncthreads();

  // causal depthwise conv (K=4) + silu -> xc
  for (int idx = t; idx < LL * DI; idx += 256) {
    const int l = idx / DI, ch = idx % DI;
    float acc = conv_b[i * DI + ch];
    const float* __restrict__ cw = conv_w + ((size_t)i * DI + ch) * KK;
#pragma unroll
    for (int k = 0; k < KK; ++k) {
      const int l2 = l - (KK - 1) + k;
      if (l2 >= 0) acc += cw[k] * s_xm[l2 * DI + ch];
    }
    s_xc[idx] = acc / (1.f + expf(-acc));
  }
  __syncthreads();

  // GEMM2: dbc = xc @ xp_w^T  (N=44, padded lda 48)
  gemm_xwt(s_xc, DI, xp_w + (size_t)i * (RR + 2 * SS) * DI, DI, LL, RR + 2 * SS, DI,
           [&](int m, int n, float v) { s_dbc[m * 48 + n] = v; });
  __syncthreads();

  // GEMM3: dt = softplus(dbc[:, :12] @ dt_w^T + dt_b) -> overwrite s_xm
  gemm_xwt(s_dbc, 48, dt_w + (size_t)i * DI * RR, RR, LL, DI, RR,
           [&](int m, int n, float v) {
             const float u = v + dt_b[i * DI + n];
             s_xm[m * DI + n] = (u > 20.f) ? u : log1pf(expf(u));
           });
  __syncthreads();

  // selective scan, state in registers; result y overwrites s_xm
  {
    float h0[SS], h1[SS];
#pragma unroll
    for (int s = 0; s < SS; ++s) { h0[s] = 0.f; h1[s] = 0.f; }
    const int ch0 = t, ch1 = t + 256;
    const bool has1 = (ch1 < DI);
    const float dsk0 = Dskip[i * DI + ch0];
    const float dsk1 = has1 ? Dskip[i * DI + ch1] : 0.f;
    for (int l = 0; l < LL; ++l) {
      const float* __restrict__ Bt = s_dbc + l * 48 + RR;
      const float* __restrict__ Ct = s_dbc + l * 48 + RR + SS;
      {
        const float dt = s_xm[l * DI + ch0];
        const float xv = s_xc[l * DI + ch0];
        const float dtx = dt * xv;
        float y = 0.f;
#pragma unroll
        for (int s = 0; s < SS; ++s) {
          const float dA = expf(dt * s_An[ch0 * SS + s]);
          h0[s] = dA * h0[s] + dtx * Bt[s];
          y += h0[s] * Ct[s];
        }
        const float zz = s_z[l * DI + ch0];
        s_xm[l * DI + ch0] = (y + dsk0 * xv) * (zz / (1.f + expf(-zz)));
      }
      if (has1) {
        const float dt = s_xm[l * DI + ch1];
        const float xv = s_xc[l * DI + ch1];
        const float dtx = dt * xv;
        float y = 0.f;
#pragma unroll
        for (int s = 0; s < SS; ++s) {
          const float dA = expf(dt * s_An[ch1 * SS + s]);
          h1[s] = dA * h1[s] + dtx * Bt[s];
          y += h1[s] * Ct[s];
        }
        const float zz = s_z[l * DI + ch1];
        s_xm[l * DI + ch1] = (y + dsk1 * xv) * (zz / (1.f + expf(-zz)));
      }
    }
  }
  __syncthreads();

  // GEMM4: tok = y @ out_w^T -> s_seq (32x192)
  gemm_xwt(s_xm, DI, out_w + (size_t)i * CC * DI, DI, LL, CC, DI,
           [&](int m, int n, float v) { s_seq[m * CC + n] = v; });
  __syncthreads();

  // GEMM5: Z = tok @ up_w^T + up_b -> LDS (reuse s_xc) + global
  float* s_Z = s_xc;
  gemm_xwt(s_seq, CC, up_w + (size_t)i * CC * CC, CC, LL, CC, CC,
           [&](int m, int n, float v) {
             const float zv = v + up_b[i * CC + n];
             s_Z[m * CC + n] = zv;
             Zg[(((size_t)i * BB + b) * CC + n) * LL + m] = zv;
           });
  __syncthreads();

  if (t < CC) {
    float s = 0.f;
    for (int l = 0; l < LL; ++l) s += s_Z[l * CC + t];
    Zmean[((size_t)i * BB + b) * CC + t] = s * (1.f / LL);
  }
}

// ---------------------------------------------------------------------------
// Kernel 3: gating + po projection + per-axis stats.  grid = 6 blocks (j,b).
// ---------------------------------------------------------------------------
__global__ void k_gate(const float* __restrict__ Zg, const float* __restrict__ Zmean,
                       const float* __restrict__ Wg_w, const float* __restrict__ Wg_b,
                       const float* __restrict__ Ws_w, const float* __restrict__ Ws_b,
                       const float* __restrict__ Wm_w, const float* __restrict__ Wm_b,
                       const float* __restrict__ po_w,
                       float* __restrict__ Pg, float* __restrict__ Psum,
                       float* __restrict__ Psq) {
  extern __shared__ float sm[];
  float* sZt  = sm;             // 6144 : Z^T (32x192)
  float* sA2  = sm + 6144;      // 6144
  float* sY   = sm + 12288;     // 6144
  float* sctx = sm + 18432;     // 192
  float* sSum = sm + 18624;     // 192
  float* sSq  = sm + 18816;     // 192
  const int j = blockIdx.x >> 1;
  const int b = blockIdx.x & 1;
  const int o1 = (j == 0) ? 1 : 0;
  const int o2 = (j == 2) ? 1 : 2;
  const int t = threadIdx.x;

  for (int idx = t; idx < LL * CC; idx += 256) {
    const int l = idx / CC, c = idx % CC;
    sZt[idx] = Zg[(((size_t)j * BB + b) * CC + c) * LL + l];
  }
  if (t < CC) {
    float acc = Wg_b[t];
    const float* __restrict__ w = Wg_w + (size_t)t * (2 * CC);
    const float* __restrict__ g1 = Zmean + ((size_t)o1 * BB + b) * CC;
    const float* __restrict__ g2 = Zmean + ((size_t)o2 * BB + b) * CC;
    for (int k = 0; k < CC; ++k) acc += w[k] * g1[k];
    for (int k = 0; k < CC; ++k) acc += w[CC + k] * g2[k];
    sctx[t] = acc;
    sSum[t] = 0.f;
    sSq[t] = 0.f;
  }
  __syncthreads();

  // a = relu(ctx + Ws@Zi + Ws_b)
  gemm_xwt(sZt, CC, Ws_w, CC, LL, CC, CC,
           [&](int m, int n, float v) {
             const float r = v + Ws_b[n] + sctx[n];
             sA2[m * CC + n] = r > 0.f ? r : 0.f;
           });
  __syncthreads();

  // Y = sigmoid(Wm@a + Wm_b) * Zi
  gemm_xwt(sA2, CC, Wm_w, CC, LL, CC, CC,
           [&](int m, int n, float v) {
             const float sg = 1.f / (1.f + expf(-(v + Wm_b[n])));
             sY[m * CC + n] = sg * sZt[m * CC + n];
           });
  __syncthreads();

  // P[j] = po_w[:, j*C:(j+1)*C] @ Y  (+ per-column sum / sumsq for stats)
  gemm_xwt(sY, CC, po_w + (size_t)j * CC, 3 * CC, LL, CC, CC,
           [&](int m, int n, float v) {
             Pg[(((size_t)j * BB + b) * CC + n) * LL + m] = v;
             atomicAdd(&sSum[n], v);
             atomicAdd(&sSq[n], v * v);
           });
  __syncthreads();

  if (t < CC) {
    Psum[((size_t)j * BB + b) * CC + t] = sSum[t];
    Psq [((size_t)j * BB + b) * CC + t] = sSq[t];
  }
}

// ---------------------------------------------------------------------------
// Kernel 3b: combine separable instance-norm statistics.
// ---------------------------------------------------------------------------
__global__ void k_stats(const float* __restrict__ Psum, const float* __restrict__ Psq,
                        float* __restrict__ muArr, float* __restrict__ rstdArr) {
  const int t = threadIdx.x;   // 384 = B*C
  if (t < BB * CC) {
    float m = 0.f, var = 0.f;
#pragma unroll
    for (int j = 0; j < 3; ++j) {
      const float s = Psum[j * (BB * CC) + t] * (1.f / LL);
      const float q = Psq [j * (BB * CC) + t] * (1.f / LL);
      m += s;
      var += q - s * s;
    }
    muArr[t] = m;                       // po_b cancels in (y - mu)
    rstdArr[t] = rsqrtf(var + 1e-5f);
  }
}

// ---------------------------------------------------------------------------
// Kernel 4: out = x + res_scale * (Pa[d]+Pc[h]+Ps[w]-mu) * rstd
// grid = B*C blocks, float4-vectorized streaming pass.
// ---------------------------------------------------------------------------
__global__ void k_final(const float* __restrict__ x, const float* __restrict__ Pg,
                        const float* __restrict__ muArr, const float* __restrict__ rstdArr,
                        const float* __restrict__ res_scale, float* __restrict__ out) {
  __shared__ float sPa[32], sPc[32], sPs[32];
  const int bc = blockIdx.x;
  const int b = bc / CC, c = bc % CC;
  const int t = threadIdx.x;
  if (t < 96) {
    const int j = t >> 5, l = t & 31;
    const float v = Pg[(((size_t)j * BB + b) * CC + c) * LL + l];
    if (j == 0) sPa[l] = v; else if (j == 1) sPc[l] = v; else sPs[l] = v;
  }
  __syncthreads();
  const float m = muArr[bc];
  const float scale = res_scale[0] * rstdArr[bc];
  const float4* __restrict__ xin = (const float4*)(x + (size_t)bc * VOL);
  float4* __restrict__ po = (float4*)(out + (size_t)bc * VOL);
  for (int q = t; q < VOL / 4; q += 256) {
    const int idx = q * 4;
    const int w = idx & 31, h = (idx >> 5) & 31, d = idx >> 10;
    const float4 xv = xin[q];
    const float base = sPa[d] + sPc[h] - m;
    float4 ov;
    ov.x = xv.x + scale * (base + sPs[w]);
    ov.y = xv.y + scale * (base + sPs[w + 1]);
    ov.z = xv.z + scale * (base + sPs[w + 2]);
    ov.w = xv.w + scale * (base + sPs[w + 3]);
    po[q] = ov;
  }
}

// ---------------------------------------------------------------------------
extern "C" void kernel_launch(void* const* d_in, const int* in_sizes, int n_in,
                              void* d_out, int out_size, void* d_ws, size_t ws_size,
                              hipStream_t stream) {
  (void)in_sizes; (void)n_in; (void)out_size; (void)ws_size;
  const float* x      = (const float*)d_in[0];
  const float* ln_w   = (const float*)d_in[1];
  const float* ln_b   = (const float*)d_in[2];
  const float* in_w   = (const float*)d_in[3];
  const float* conv_w = (const float*)d_in[4];
  const float* conv_b = (const float*)d_in[5];
  const float* xp_w   = (const float*)d_in[6];
  const float* dt_w   = (const float*)d_in[7];
  const float* dt_b   = (const float*)d_in[8];
  const float* A_log  = (const float*)d_in[9];
  const float* Dskip  = (const float*)d_in[10];
  const float* out_w  = (const float*)d_in[11];
  const float* up_w   = (const float*)d_in[12];
  const float* up_b   = (const float*)d_in[13];
  const float* Wg_w   = (const float*)d_in[14];
  const float* Wg_b   = (const float*)d_in[15];
  const float* Ws_w   = (const float*)d_in[16];
  const float* Ws_b   = (const float*)d_in[17];
  const float* Wm_w   = (const float*)d_in[18];
  const float* Wm_b   = (const float*)d_in[19];
  const float* po_w   = (const float*)d_in[20];
  const float* po_b   = (const float*)d_in[21];  // cancels analytically; unused
  const float* rscale = (const float*)d_in[22];
  (void)po_b;

  float* ws    = (float*)d_ws;
  float* seqs  = ws;            // 3*B*C*32 = 36864
  float* Zg    = ws + 36864;    // 36864
  float* Zmean = ws + 73728;    // 1152
  float* Pg    = ws + 74880;    // 36864
  float* Psum  = ws + 111744;   // 1152
  float* Psq   = ws + 112896;   // 1152
  float* muA   = ws + 114048;   // 384
  float* rstd  = ws + 114432;   // 384

  k_axis_means<<<BB * CC, 256, 0, stream>>>(x, seqs);

  const size_t sm2 = 50688u * sizeof(float);   // ~203 KB LDS (<=320 KB/WGP)
  k_branch<<<6, 256, sm2, stream>>>(seqs, ln_w, ln_b, in_w, conv_w, conv_b,
                                    xp_w, dt_w, dt_b, A_log, Dskip, out_w,
                                    up_w, up_b, Zg, Zmean);

  const size_t sm3 = 19008u * sizeof(float);
  k_gate<<<6, 256, sm3, stream>>>(Zg, Zmean, Wg_w, Wg_b, Ws_w, Ws_b,
                                  Wm_w, Wm_b, po_w, Pg, Psum, Psq);

  k_stats<<<1, 384, 0, stream>>>(Psum, Psq, muA, rstd);

  k_final<<<BB * CC, 256, 0, stream>>>(x, Pg, muA, rstd, rscale, (float*)d_out);
}